// GlmMLA_39874476376635
// MI455X (gfx1250) — compile-verified
//
#include <hip/hip_runtime.h>

// Problem constants (match reference)
#define B_    32
#define S_    2048
#define H_    32
#define NOPE_ 128
#define ROPE_ 64
#define VD_   128
#define HID_  4096
#define QHD_  192   // NOPE + ROPE

typedef __attribute__((ext_vector_type(2))) float v2f;
typedef __attribute__((ext_vector_type(8))) float v8f;

// ---------------------------------------------------------------------------
// 32x16 f32 output tile (two 16x16 accumulators sharing each B fragment) via
// V_WMMA_F32_16X16X4_F32. Each weight element is loaded exactly once.
// Per-lane layout (ISA 7.12.2, 32-bit 16x4 A / 4x16 B):
//   lanes 0-15 take K-pair {k,k+1} of row lmod; lanes 16-31 take {k+2,k+3}
//   -> pointers pre-offset by +2*(lane>>4).
// ---------------------------------------------------------------------------
__device__ __forceinline__ void wmma_gemm_2m(const float* __restrict__ Arow0,
                                             const float* __restrict__ Arow1,
                                             const float* __restrict__ Brow,
                                             int K, v8f& acc0, v8f& acc1) {
#pragma unroll 8
    for (int k = 0; k < K; k += 4) {
        v2f b  = *(const v2f*)(Brow + k);
        v2f a0 = *(const v2f*)(Arow0 + k);
        v2f a1 = *(const v2f*)(Arow1 + k);
        acc0 = __builtin_amdgcn_wmma_f32_16x16x4_f32(false, a0, false, b,
                                                     (short)0, acc0, false, false);
        acc1 = __builtin_amdgcn_wmma_f32_16x16x4_f32(false, a1, false, b,
                                                     (short)0, acc1, false, false);
    }
}

// D layout: VGPR r, lanes 0-15 -> (m=r, n=lane); lanes 16-31 -> (m=r+8, n=lane-16).
__device__ __forceinline__ void store_tile(float* __restrict__ dst, int ldd,
                                           v8f acc, int lmod, int lhalf) {
    float* p = dst + (size_t)(lhalf * 8) * ldd + lmod;
#pragma unroll
    for (int r = 0; r < 8; ++r) p[(size_t)r * ldd] = acc[r];
}

// ---------------------------------------------------------------------------
// Kernel 1: fused QKV projection (M = 32 fully covered per wave).
//  N-space: [0,4096)     -> q_nope (Wq rows head*192 + j, j<128; ROPE skipped)
//           [4096,12288) -> k then v (Wkv rows n-4096)
// q_ws: [32][4096]   kv_ws: [32][8192] (k | v concatenated)
// 768 N-tiles -> 768 waves -> 96 blocks x 8 waves.
// ---------------------------------------------------------------------------
__global__ __launch_bounds__(256) void qkv_kernel(const float* __restrict__ X,
                                                  const float* __restrict__ Wq,
                                                  const float* __restrict__ Wkv,
                                                  float* __restrict__ q_ws,
                                                  float* __restrict__ kv_ws) {
    const int wave  = blockIdx.x * (blockDim.x >> 5) + (threadIdx.x >> 5);
    const int lane  = threadIdx.x & 31;
    const int lmod  = lane & 15;
    const int lhalf = lane >> 4;

    const int n0 = wave << 4;   // 0..12272

    const float* W;
    int    wrow0;
    float* dst;
    int    ldd;
    if (n0 < H_ * NOPE_) {                 // q_nope segment
        const int head = n0 >> 7;
        wrow0 = head * QHD_ + (n0 & 127);  // skip ROPE columns of Wq
        W     = Wq;
        dst   = q_ws + n0;
        ldd   = H_ * NOPE_;
    } else {                               // k / v segment (shared row mapping)
        wrow0 = n0 - H_ * NOPE_;           // 0..8191 into Wkv
        W     = Wkv;
        dst   = kv_ws + wrow0;
        ldd   = 8192;
    }

    const float* Arow0 = X + (size_t)lmod * HID_ + 2 * lhalf;
    const float* Arow1 = X + (size_t)(16 + lmod) * HID_ + 2 * lhalf;
    const float* Brow  = W + (size_t)(wrow0 + lmod) * HID_ + 2 * lhalf;

    v8f acc0 = {0.f, 0.f, 0.f, 0.f, 0.f, 0.f, 0.f, 0.f};
    v8f acc1 = acc0;
    wmma_gemm_2m(Arow0, Arow1, Brow, HID_, acc0, acc1);

    store_tile(dst, ldd, acc0, lmod, lhalf);                       // rows 0..15
    store_tile(dst + (size_t)16 * ldd, ldd, acc1, lmod, lhalf);    // rows 16..31
}

// ---------------------------------------------------------------------------
// Kernel 2: decode attention, one (b,h) per block (1024 blocks, 256 thr).
// Score phase: one wave per position, lanes cover dims with float4 loads
// (fully coalesced 512B per K row) + shfl_xor wave reduction.
// Position seq_len-1 substitutes the freshly projected k/v from kv_ws
// (input caches must not be mutated; matches the reference scatter+read).
// ---------------------------------------------------------------------------
__global__ __launch_bounds__(256) void attn_kernel(const float* __restrict__ q_ws,
                                                   const float* __restrict__ kv_ws,
                                                   const float* __restrict__ k_cache,
                                                   const float* __restrict__ v_cache,
                                                   const long long* __restrict__ seq_lens,
                                                   float* __restrict__ o_ws) {
    const int b   = blockIdx.x >> 5;   // / H
    const int h   = blockIdx.x & 31;
    const int tid = threadIdx.x;
    const int wv  = tid >> 5;          // wave in block (0..7)
    const int ln  = tid & 31;
    const int L   = (int)seq_lens[b];
    const float scale = 0.08838834764831845f;  // 128^-0.5

    __shared__ float sc[S_];      // 8 KB score buffer
    __shared__ float red[256];

    // each lane holds q dims [ln*4, ln*4+4)
    const float4 qv = *(const float4*)(q_ws + (size_t)b * (H_ * NOPE_) + h * NOPE_ + ln * 4);

    // scores: one wave per position
    for (int s = wv; s < L; s += 8) {
        const float* kp = (s == L - 1)
            ? kv_ws + (size_t)b * 8192 + h * NOPE_
            : k_cache + (((size_t)b * S_ + s) * H_ + h) * NOPE_;
        const float4 kk = *(const float4*)(kp + ln * 4);
        float p = qv.x * kk.x + qv.y * kk.y + qv.z * kk.z + qv.w * kk.w;
#pragma unroll
        for (int off = 16; off > 0; off >>= 1) p += __shfl_xor(p, off, 32);
        if (ln == 0) sc[s] = p * scale;
    }
    __syncthreads();

    // block max
    float m = -3.402823466e38f;
    for (int s = tid; s < L; s += 256) m = fmaxf(m, sc[s]);
    red[tid] = m;
    __syncthreads();
    for (int off = 128; off > 0; off >>= 1) {
        if (tid < off) red[tid] = fmaxf(red[tid], red[tid + off]);
        __syncthreads();
    }
    m = red[0];
    __syncthreads();

    // exp + block sum
    float ssum = 0.f;
    for (int s = tid; s < L; s += 256) {
        float e = __expf(sc[s] - m);
        sc[s]   = e;
        ssum   += e;
    }
    red[tid] = ssum;
    __syncthreads();
    for (int off = 128; off > 0; off >>= 1) {
        if (tid < off) red[tid] += red[tid + off];
        __syncthreads();
    }
    const float inv = 1.0f / red[0];
    __syncthreads();

    // weighted V: two thread-groups split s; 128 lanes cover dims (coalesced)
    const int d   = tid & 127;
    const int grp = tid >> 7;
    float acc = 0.f;
    for (int s = grp; s < L; s += 2) {
        const float* vp = (s == L - 1)
            ? kv_ws + (size_t)b * 8192 + 4096 + h * VD_
            : v_cache + (((size_t)b * S_ + s) * H_ + h) * VD_;
        acc = fmaf(sc[s], vp[d], acc);
    }
    red[tid] = acc;
    __syncthreads();
    if (grp == 0)
        o_ws[(size_t)b * (H_ * VD_) + h * VD_ + d] = (red[tid] + red[tid + 128]) * inv;
}

// ---------------------------------------------------------------------------
// Kernel 3: output projection out[32,4096] = o_ws[32,4096] @ Wo^T.
// 256 N-tiles -> 256 waves -> 32 blocks x 8 waves.
// ---------------------------------------------------------------------------
__global__ __launch_bounds__(256) void oproj_kernel(const float* __restrict__ Aat,
                                                    const float* __restrict__ Wo,
                                                    float* __restrict__ out) {
    const int wave  = blockIdx.x * (blockDim.x >> 5) + (threadIdx.x >> 5);
    const int lane  = threadIdx.x & 31;
    const int lmod  = lane & 15;
    const int lhalf = lane >> 4;

    const int n0 = wave << 4;   // 0..4080

    const float* Arow0 = Aat + (size_t)lmod * HID_ + 2 * lhalf;
    const float* Arow1 = Aat + (size_t)(16 + lmod) * HID_ + 2 * lhalf;
    const float* Brow  = Wo + (size_t)(n0 + lmod) * HID_ + 2 * lhalf;

    v8f acc0 = {0.f, 0.f, 0.f, 0.f, 0.f, 0.f, 0.f, 0.f};
    v8f acc1 = acc0;
    wmma_gemm_2m(Arow0, Arow1, Brow, HID_, acc0, acc1);

    store_tile(out + n0, HID_, acc0, lmod, lhalf);
    store_tile(out + (size_t)16 * HID_ + n0, HID_, acc1, lmod, lhalf);
}

// ---------------------------------------------------------------------------
extern "C" void kernel_launch(void* const* d_in, const int* in_sizes, int n_in,
                              void* d_out, int out_size, void* d_ws, size_t ws_size,
                              hipStream_t stream) {
    const float*     hidden   = (const float*)d_in[0];
    // d_in[1] = positions (unused: seq_lens carries the same info)
    const float*     k_cache  = (const float*)d_in[2];
    const float*     v_cache  = (const float*)d_in[3];
    const float*     Wq       = (const float*)d_in[4];
    const float*     Wkv      = (const float*)d_in[5];
    const float*     Wo       = (const float*)d_in[6];
    // d_in[7] = slot_mapping (implied by seq_lens: slot = b*S + seq_len-1)
    const long long* seq_lens = (const long long*)d_in[8];
    float*           out      = (float*)d_out;

    float* q_ws  = (float*)d_ws;               // 32*4096 floats (512 KB)
    float* kv_ws = q_ws + B_ * (H_ * NOPE_);   // 32*8192 floats (1 MB)
    float* o_ws  = kv_ws + B_ * 8192;          // 32*4096 floats (512 KB)

    // 768 N-tile waves / 8 per block
    qkv_kernel<<<96, 256, 0, stream>>>(hidden, Wq, Wkv, q_ws, kv_ws);
    // one block per (b,h)
    attn_kernel<<<B_ * H_, 256, 0, stream>>>(q_ws, kv_ws, k_cache, v_cache,
                                             seq_lens, o_ws);
    // 256 N-tile waves / 8 per block
    oproj_kernel<<<32, 256, 0, stream>>>(o_ws, Wo, out);
}